// Explicit_corr1_55387898249792
// MI455X (gfx1250) — compile-verified
//
#include <hip/hip_runtime.h>
#include <hip/hip_bf16.h>

// ---------------------------------------------------------------------------
// Problem constants (from the reference):
//   B=64, C=256, candidate grid 20x20 (400 positions), window SIZE=4 (16 px),
//   NPOS=17 -> 289 output positions, MLP: 512->256->128->64->32->1, 3 iters.
// Workspace: P = w0[:, :256] @ x, U = w0[:, 256:] @ x per grid position.
// ---------------------------------------------------------------------------

typedef __attribute__((ext_vector_type(2))) float v2f;
typedef __attribute__((ext_vector_type(8))) float v8f;
typedef __attribute__((ext_vector_type(4))) unsigned int u32x4;
typedef __attribute__((ext_vector_type(8))) int i32x8;
typedef __attribute__((ext_vector_type(4))) int i32x4;

__device__ __forceinline__ float lrelu(float x) { return x > 0.0f ? x : 0.01f * x; }

// D = A(16x4) * B(4x16) + C, all f32 (V_WMMA_F32_16X16X4_F32).
__device__ __forceinline__ v8f wmma4(v2f a, v2f b, v8f c) {
  return __builtin_amdgcn_wmma_f32_16x16x4_f32(false, a, false, b, (short)0, c,
                                               false, false);
}

// ---------------------------------------------------------------------------
// Tensor Data Mover: 3D tile load Global -> LDS.
//   addr = global + 4*(x + y*stride0 + z*stride1), x<d0, y<d1, z<d2,
//   written to LDS contiguously in [z][y][x] order. Strides in elements (f32).
// D# bitfields per CDNA5 ISA. 6-arg clang-23 builtin:
// (g0, g1, g2, g3, extra, cpol) — extra group zeroed (unused in normal mode).
// ---------------------------------------------------------------------------
__device__ __forceinline__ void tdm_load_3d(unsigned lds_off, const void* gptr,
                                            unsigned d0, unsigned d1, unsigned d2,
                                            unsigned stride0, unsigned stride1) {
  unsigned long long ga = (unsigned long long)(size_t)gptr;
  const unsigned td = 1u << 30;  // huge tensor dims: disable OOB clamping
  u32x4 g0;
  g0[0] = 1u;                                                // count=1, user D#
  g0[1] = lds_off;                                           // LDS byte addr
  g0[2] = (unsigned)ga;                                      // global_addr lo
  g0[3] = ((unsigned)(ga >> 32) & 0x01FFFFFFu) | 0x80000000u; // addr hi | type=2
  i32x8 g1;
  g1[0] = (int)(2u << 16);                                   // data_size=4B
  g1[1] = (int)((td & 0xFFFFu) << 16);                       // tensor_dim0 lo
  g1[2] = (int)(((td >> 16) & 0xFFFFu) | ((td & 0xFFFFu) << 16));
  g1[3] = (int)(((td >> 16) & 0xFFFFu) | ((d0 & 0xFFFFu) << 16)); // tile_dim0
  g1[4] = (int)((d1 & 0xFFFFu) | ((d2 & 0xFFFFu) << 16));    // tile_dim1/2
  g1[5] = (int)stride0;                                      // dim0_stride lo
  g1[6] = (int)((stride1 & 0xFFFFu) << 16);                  // dim1_stride lo
  g1[7] = (int)(stride1 >> 16);                              // dim1_stride hi
  i32x4 g2; g2[0] = (int)td; g2[1] = 0; g2[2] = 0; g2[3] = 0; // tensor_dim2
  i32x4 g3; g3[0] = 0; g3[1] = 0; g3[2] = 0; g3[3] = 0;
  i32x8 gx; gx[0]=0; gx[1]=0; gx[2]=0; gx[3]=0; gx[4]=0; gx[5]=0; gx[6]=0; gx[7]=0;
  __builtin_amdgcn_tensor_load_to_lds(g0, g1, g2, g3, gx, 0);
}

// ---------------------------------------------------------------------------
// Kernel 1: precompute P[b][pos][c] = sum_i w0[c][i]     * x[b][i][pos]
//                      U[b][pos][c] = sum_i w0[c][256+i] * x[b][i][pos]
// GEMM (512x256) @ (256 x 16) per tile of 16 positions, f32 WMMA.
// Grid: 64 batches * (25 candidate tiles + 1 template tile), 256 threads.
// ---------------------------------------------------------------------------
__global__ __launch_bounds__(256) void pre_kernel(
    const float* __restrict__ tf, const float* __restrict__ cf,
    const float* __restrict__ w0, float* __restrict__ P, float* __restrict__ U,
    float* __restrict__ Pt, float* __restrict__ Ut) {
  extern __shared__ float xs[];  // x tile staged as [c=256][pos=16]
  const int b = blockIdx.x / 26;
  const int tile = blockIdx.x % 26;
  const bool isT = (tile == 25);
  const float* src = isT ? (tf + (size_t)b * 256 * 16)
                         : (cf + (size_t)b * 256 * 400 + tile * 16);
  const int rs = isT ? 16 : 400;
  {
    const int t = threadIdx.x;  // one channel row per thread, 16 contiguous f32
    const float4* s4 = (const float4*)(src + (size_t)t * rs);
    float4* d4 = (float4*)(xs + t * 16);
    d4[0] = s4[0]; d4[1] = s4[1]; d4[2] = s4[2]; d4[3] = s4[3];
  }
  __syncthreads();
  const int lane = threadIdx.x & 31, lr = lane & 15, lh = lane >> 4;
  const int wv = threadIdx.x >> 5;
  for (int sub = 0; sub < 4; ++sub) {       // 32 M-tiles over 8 waves
    const int mt = wv + sub * 8;
    const bool isU = mt >= 16;              // rows 256..511 -> U (w0[:,256:])
    const int row = (isU ? mt - 16 : mt) * 16 + lr;
    const float* wb = w0 + (size_t)row * 512 + (isU ? 256 : 0) + 2 * lh;
    v8f acc = {0, 0, 0, 0, 0, 0, 0, 0};
#pragma unroll 4
    for (int kk = 0; kk < 64; ++kk) {       // K = 256, 4 per WMMA
      v2f a = *(const v2f*)(wb + kk * 4);
      const int kx = kk * 4 + 2 * lh;
      v2f bb; bb.x = xs[kx * 16 + lr]; bb.y = xs[kx * 16 + 16 + lr];
      acc = wmma4(a, bb, acc);
    }
    const size_t pixbase = isT ? ((size_t)b * 16 + lr) * 256
                               : ((size_t)b * 400 + tile * 16 + lr) * 256;
    float* dst = (isU ? (isT ? Ut : U) : (isT ? Pt : P)) + pixbase + (mt & 15) * 16;
#pragma unroll
    for (int v = 0; v < 8; ++v) dst[v + 8 * lh] = acc[v];  // D: M=v+8*lh, N=lr
  }
}

// ---------------------------------------------------------------------------
// Kernel 2: one workgroup = one tile (b, oi, oj) of 16 window pixels (or one
// template batch). 3 reweighting iterations fully local; softmax over the 16
// pixels. MLP layers 256->128->64->32 as f32 WMMA, weights from L2-resident
// global, activations via LDS. P/U/X tiles staged by TDM.
// ---------------------------------------------------------------------------
__global__ __launch_bounds__(256) void corr_kernel(
    const float* __restrict__ tf, const float* __restrict__ cf,
    const float* __restrict__ w1, const float* __restrict__ w2,
    const float* __restrict__ w3, const float* __restrict__ w4,
    const float* __restrict__ P, const float* __restrict__ U,
    const float* __restrict__ Pt, const float* __restrict__ Ut,
    float* __restrict__ out_ct, float* __restrict__ out_cs,
    float* __restrict__ out_tw, float* __restrict__ out_sw) {
  extern __shared__ float smem[];
  float* Pl = smem;           // [pix=16][c=256]  (TDM layout z=py,y=px,x=c)
  float* Ul = smem + 4096;    // [pix=16][c=256]
  float* Xl = smem + 8192;    // [c=256][pix=16]  (TDM layout z=c,y=py,x=px)
  float* h0 = smem + 12288;   // [k=256][pix=16]
  float* h1 = smem + 16384;   // [k=128][pix=16]
  float* h2 = smem + 18432;   // [k=64][pix=16]
  float* h3 = smem + 19456;   // [k=32][pix=16]
  float* w4s = smem + 19968;  // 32
  float* sacc = smem + 20000; // running weight s per pixel, 16
  float* logit = smem + 20016;// 16

  const int bid = blockIdx.x;
  const bool isT = bid >= 64 * 289;
  int b, oidx = 0;
  if (isT) { b = bid - 64 * 289; } else { b = bid / 289; oidx = bid % 289; }
  const int oi = oidx / 17, oj = oidx % 17;
  const int tid = threadIdx.x;

  if ((tid >> 5) == 0) {  // wave 0 drives the TDM; others wait at the barrier
    const unsigned oP = (unsigned)(size_t)(void*)Pl;
    const unsigned oU = (unsigned)(size_t)(void*)Ul;
    const unsigned oX = (unsigned)(size_t)(void*)Xl;
    if (!isT) {
      const size_t pb = ((size_t)b * 400 + oi * 20 + oj) * 256;
      tdm_load_3d(oP, P + pb, 256, 4, 4, 256, 5120);   // x=c, y=px, z=py
      tdm_load_3d(oU, U + pb, 256, 4, 4, 256, 5120);
      tdm_load_3d(oX, cf + (size_t)b * 102400 + oi * 20 + oj, 4, 4, 256, 20, 400);
    } else {
      tdm_load_3d(oP, Pt + (size_t)b * 4096, 4096, 1, 1, 4096, 4096);
      tdm_load_3d(oU, Ut + (size_t)b * 4096, 4096, 1, 1, 4096, 4096);
      tdm_load_3d(oX, tf + (size_t)b * 4096, 4096, 1, 1, 4096, 4096);
    }
    __builtin_amdgcn_s_wait_tensorcnt(0);
  }
  if (tid >= 32 && tid < 64) w4s[tid - 32] = w4[tid - 32];
  if (tid >= 64 && tid < 80) sacc[tid - 64] = 1.0f;
  __syncthreads();

  const int lane = tid & 31, lr = lane & 15, lh = lane >> 4;
  const int wv = tid >> 5;

  for (int it = 0; it < 3; ++it) {
    {  // v_k = (1/16) sum_p s_p * P[p][k];  h0[k][p] = lrelu(v_k + s_p*U[p][k])
      float s_[16];
#pragma unroll
      for (int p = 0; p < 16; ++p) s_[p] = sacc[p];
      float vk = 0.f;
#pragma unroll
      for (int p = 0; p < 16; ++p) vk += s_[p] * Pl[p * 256 + tid];
      vk *= 0.0625f;
#pragma unroll
      for (int p = 0; p < 16; ++p)
        h0[tid * 16 + p] = lrelu(vk + s_[p] * Ul[p * 256 + tid]);
    }
    __syncthreads();
    {  // layer 2: 256 -> 128, 8 waves x (M=16) x 64 K-steps
      v8f acc = {0, 0, 0, 0, 0, 0, 0, 0};
      const float* wb = w1 + (size_t)(wv * 16 + lr) * 256 + 2 * lh;
#pragma unroll 4
      for (int kk = 0; kk < 64; ++kk) {
        v2f a = *(const v2f*)(wb + kk * 4);
        const int kx = kk * 4 + 2 * lh;
        v2f bb; bb.x = h0[kx * 16 + lr]; bb.y = h0[kx * 16 + 16 + lr];
        acc = wmma4(a, bb, acc);
      }
#pragma unroll
      for (int v = 0; v < 8; ++v)
        h1[(wv * 16 + v + 8 * lh) * 16 + lr] = lrelu(acc[v]);
    }
    __syncthreads();
    if (wv < 4) {  // layer 3: 128 -> 64
      v8f acc = {0, 0, 0, 0, 0, 0, 0, 0};
      const float* wb = w2 + (size_t)(wv * 16 + lr) * 128 + 2 * lh;
#pragma unroll 4
      for (int kk = 0; kk < 32; ++kk) {
        v2f a = *(const v2f*)(wb + kk * 4);
        const int kx = kk * 4 + 2 * lh;
        v2f bb; bb.x = h1[kx * 16 + lr]; bb.y = h1[kx * 16 + 16 + lr];
        acc = wmma4(a, bb, acc);
      }
#pragma unroll
      for (int v = 0; v < 8; ++v)
        h2[(wv * 16 + v + 8 * lh) * 16 + lr] = lrelu(acc[v]);
    }
    __syncthreads();
    if (wv < 2) {  // layer 4: 64 -> 32
      v8f acc = {0, 0, 0, 0, 0, 0, 0, 0};
      const float* wb = w3 + (size_t)(wv * 16 + lr) * 64 + 2 * lh;
#pragma unroll 4
      for (int kk = 0; kk < 16; ++kk) {
        v2f a = *(const v2f*)(wb + kk * 4);
        const int kx = kk * 4 + 2 * lh;
        v2f bb; bb.x = h2[kx * 16 + lr]; bb.y = h2[kx * 16 + 16 + lr];
        acc = wmma4(a, bb, acc);
      }
#pragma unroll
      for (int v = 0; v < 8; ++v)
        h3[(wv * 16 + v + 8 * lh) * 16 + lr] = lrelu(acc[v]);
    }
    __syncthreads();
    if (tid < 16) {  // layer 5: 32 -> 1 logit per pixel
      float l = 0.f;
#pragma unroll
      for (int k = 0; k < 32; ++k) l += w4s[k] * h3[k * 16 + tid];
      logit[tid] = l;
    }
    __syncthreads();
    if (tid < 16) {  // softmax over the 16 window pixels; fold into s
      float mx = logit[0];
#pragma unroll
      for (int p = 1; p < 16; ++p) mx = fmaxf(mx, logit[p]);
      float den = 0.f;
#pragma unroll
      for (int p = 0; p < 16; ++p) den += __expf(logit[p] - mx);
      const float wn = __expf(logit[tid] - mx) / den;
      sacc[tid] *= wn;
      if (it == 2) {  // last iteration's wn is the weight output
        if (!isT) out_sw[((size_t)b * 16 + tid) * 289 + oidx] = wn;
        else      out_tw[b * 16 + tid] = wn;
      }
    }
    __syncthreads();
  }
  {  // common = sum_p x[c][p] * s_final[p]
    float a = 0.f;
#pragma unroll
    for (int p = 0; p < 16; ++p) a += Xl[tid * 16 + p] * sacc[p];
    if (!isT) out_cs[((size_t)b * 256 + tid) * 289 + oidx] = a;
    else      out_ct[b * 256 + tid] = a;
  }
}

extern "C" void kernel_launch(void* const* d_in, const int* in_sizes, int n_in,
                              void* d_out, int out_size, void* d_ws, size_t ws_size,
                              hipStream_t stream) {
  const float* tf = (const float*)d_in[0];  // (64,256,4,4)
  const float* cf = (const float*)d_in[1];  // (64,256,20,20)
  const float* w0 = (const float*)d_in[2];  // (256,512)
  const float* w1 = (const float*)d_in[3];  // (128,256)
  const float* w2 = (const float*)d_in[4];  // (64,128)
  const float* w3 = (const float*)d_in[5];  // (32,64)
  const float* w4 = (const float*)d_in[6];  // (1,32)

  float* P  = (float*)d_ws;                        // 64*400*256
  float* U  = P + (size_t)64 * 400 * 256;          // 64*400*256
  float* Pt = U + (size_t)64 * 400 * 256;          // 64*16*256
  float* Ut = Pt + (size_t)64 * 16 * 256;          // 64*16*256

  float* out   = (float*)d_out;
  float* o_ct  = out;                              // (64,256,1,1)
  float* o_cs  = out + 16384;                      // (64,256,1,1,17,17)
  float* o_tw  = out + 16384 + 4734976;            // (64,1,4,4)
  float* o_sw  = out + 16384 + 4734976 + 1024;     // (64,1,4,4,17,17)

  pre_kernel<<<dim3(64 * 26), dim3(256), 256 * 16 * 4, stream>>>(
      tf, cf, w0, P, U, Pt, Ut);
  corr_kernel<<<dim3(64 * 289 + 64), dim3(256), 20032 * 4, stream>>>(
      tf, cf, w1, w2, w3, w4, P, U, Pt, Ut, o_ct, o_cs, o_tw, o_sw);
}